// SSITrimLoss_62594853372199
// MI455X (gfx1250) — compile-verified
//
#include <hip/hip_runtime.h>
#include <hip/hip_bf16.h>
#include <stdint.h>

#define NBINS 65536
#define BATCH 32   // samples

typedef __attribute__((ext_vector_type(2))) float v2f;
typedef __attribute__((ext_vector_type(8))) float v8f;

// ---------------------------------------------------------------------------
// Async global->LDS helpers (gfx1250: GLOBAL_LOAD_ASYNC_TO_LDS_*, ASYNCcnt)
// ---------------------------------------------------------------------------
__device__ __forceinline__ void async_ld16(uint32_t lds_off, const void* gptr) {
    unsigned long long ga = (unsigned long long)(uintptr_t)gptr;
    asm volatile("global_load_async_to_lds_b128 %0, %1, off"
                 :: "v"(lds_off), "v"(ga) : "memory");
}
__device__ __forceinline__ void wait_async_le3() {
    asm volatile("s_wait_asynccnt 0x3" ::: "memory");
}
__device__ __forceinline__ void wait_async_0() {
    asm volatile("s_wait_asynccnt 0x0" ::: "memory");
}
__device__ __forceinline__ void wait_ds_0() {
    asm volatile("s_wait_dscnt 0x0" ::: "memory");
}

// ---------------------------------------------------------------------------
// Wave32 sum via the matrix unit. A = ones (16x4) is layout-independent; B's
// two VGPRs carry {v, 0} per lane, so sum(all B entries) == sum over lanes.
// Every row of D = colsum[N]; D VGPR0 holds colsum for rows M=0 (lanes 0-15)
// and M=8 (lanes 16-31), so a 4-step xor-shuffle over 16-lane groups yields
// the full wave sum in every lane. Requires EXEC == all ones (call only at
// uniform convergence points).
// ---------------------------------------------------------------------------
__device__ __forceinline__ float wave_sum_wmma(float v) {
    v2f a = {1.f, 1.f};
    v2f b = {v, 0.f};
    v8f c = {};
    c = __builtin_amdgcn_wmma_f32_16x16x4_f32(false, a, false, b,
                                              (short)0, c, false, false);
    float s = c[0];
#pragma unroll
    for (int o = 8; o > 0; o >>= 1) s += __shfl_xor(s, o, 32);
    return s;
}

__device__ __forceinline__ float wave_sum_shfl(float v) {
#pragma unroll
    for (int o = 16; o > 0; o >>= 1) v += __shfl_xor(v, o, 32);
    return v;
}

// ---------------------------------------------------------------------------
// Pass 1: masked moments per sample: n, sum(p), sum(g), sum(p*p), sum(p*g)
// Async double-buffered LDS staging; WMMA wave reduction -> f64 atomics.
// Grid: dim3(bps, BATCH).
// ---------------------------------------------------------------------------
__global__ void __launch_bounds__(256)
k_stats(const float4* __restrict__ pred, const float4* __restrict__ gt,
        const int4* __restrict__ mask, double* __restrict__ dstats, int N4) {
    __shared__ float4 sP[2][8][32];
    __shared__ float4 sG[2][8][32];
    __shared__ int4   sM[2][8][32];

    const int b    = blockIdx.y;
    const int sblk = blockIdx.x;
    const int wave = threadIdx.x >> 5;
    const int lane = threadIdx.x & 31;
    const size_t base   = (size_t)b * (size_t)N4;
    const int    stride = gridDim.x * 256;
    const int    tid    = sblk * 256 + threadIdx.x;
    const int    iters  = N4 / stride;   // uniform across all lanes

    uint32_t oP[2], oG[2], oM[2];
    oP[0] = (uint32_t)(uintptr_t)&sP[0][wave][lane];
    oP[1] = (uint32_t)(uintptr_t)&sP[1][wave][lane];
    oG[0] = (uint32_t)(uintptr_t)&sG[0][wave][lane];
    oG[1] = (uint32_t)(uintptr_t)&sG[1][wave][lane];
    oM[0] = (uint32_t)(uintptr_t)&sM[0][wave][lane];
    oM[1] = (uint32_t)(uintptr_t)&sM[1][wave][lane];

    float nl = 0.f, spl = 0.f, sgl = 0.f, sppl = 0.f, spgl = 0.f;

    auto issue = [&](int buf, int j) {
        async_ld16(oP[buf], pred + base + j);
        async_ld16(oG[buf], gt   + base + j);
        async_ld16(oM[buf], mask + base + j);
    };

    if (iters > 0) issue(0, tid);
    for (int i = 0; i < iters; ++i) {
        if (i + 1 < iters) {
            wait_ds_0();                       // prior reads of target buffer done
            issue((i + 1) & 1, tid + (i + 1) * stride);
            wait_async_le3();                  // current batch (3 ops) complete
        } else {
            wait_async_0();
        }
        float4 p = sP[i & 1][wave][lane];
        float4 g = sG[i & 1][wave][lane];
        int4   m = sM[i & 1][wave][lane];
#pragma unroll
        for (int c = 0; c < 4; ++c) {
            float pc = (&p.x)[c], gc = (&g.x)[c];
            float mf = ((&m.x)[c] != 0) ? 1.f : 0.f;
            nl   += mf;
            spl  += pc * mf;
            sgl  += gc * mf;
            sppl += pc * pc * mf;
            spgl += pc * gc * mf;
        }
    }
    // remainder (generic shapes); plain loads
    int j = tid + iters * stride;
    if (j < N4) {
        float4 p = pred[base + j];
        float4 g = gt[base + j];
        int4   m = mask[base + j];
#pragma unroll
        for (int c = 0; c < 4; ++c) {
            float pc = (&p.x)[c], gc = (&g.x)[c];
            float mf = ((&m.x)[c] != 0) ? 1.f : 0.f;
            nl += mf; spl += pc * mf; sgl += gc * mf;
            sppl += pc * pc * mf; spgl += pc * gc * mf;
        }
    }

    // EXEC is all-ones here (uniform convergence) -> WMMA reductions legal.
    nl   = wave_sum_wmma(nl);
    spl  = wave_sum_wmma(spl);
    sgl  = wave_sum_wmma(sgl);
    sppl = wave_sum_wmma(sppl);
    spgl = wave_sum_wmma(spgl);
    if (lane == 0) {
        double* s = dstats + b * 8;
        atomicAdd(&s[0], (double)nl);
        atomicAdd(&s[1], (double)spl);
        atomicAdd(&s[2], (double)sgl);
        atomicAdd(&s[3], (double)sppl);
        atomicAdd(&s[4], (double)spgl);
    }
}

// ---------------------------------------------------------------------------
// alpha/beta/k per sample
// ---------------------------------------------------------------------------
__global__ void k_params(const double* __restrict__ dstats, unsigned* __restrict__ prm) {
    int b = threadIdx.x;
    if (b >= BATCH) return;
    const double* s = dstats + b * 8;
    float nf = (float)s[0];
    float ns = fmaxf(nf, 1.f);
    float md = (float)s[1] / ns;
    float mz = (float)s[2] / ns;
    float vard = (float)s[3] / ns - md * md;
    float cov  = (float)s[4] / ns - md * mz;
    float alpha = cov / (vard + 1e-6f);
    float beta  = mz - alpha * md;
    int k = (int)floorf((1.0f - 0.2f) * nf);
    unsigned* pr = prm + b * 8;
    pr[0] = __float_as_uint(alpha);
    pr[1] = __float_as_uint(beta);
    pr[2] = (unsigned)(k > 0 ? k : 0);
}

// ---------------------------------------------------------------------------
// Residual passes. mode 0: hist of key>>16. mode 1: hist of key&0xFFFF where
// key>>16 == T. mode 2: sum residuals with key < K (WMMA wave sum -> f64).
// Grid: dim3(bps, BATCH).
// ---------------------------------------------------------------------------
__global__ void __launch_bounds__(256)
k_resid(const float4* __restrict__ pred, const float4* __restrict__ gt,
        const int4* __restrict__ mask, const unsigned* __restrict__ prm,
        unsigned* __restrict__ hist, double* __restrict__ dstats,
        int N4, int mode) {
    const int b    = blockIdx.y;
    const int sblk = blockIdx.x;
    const unsigned* pr = prm + b * 8;
    const float alpha = __uint_as_float(pr[0]);
    const float beta  = __uint_as_float(pr[1]);
    const unsigned T  = pr[3];
    const unsigned K  = pr[5];
    unsigned* h = hist + (size_t)b * NBINS;
    const size_t base = (size_t)b * (size_t)N4;
    const int stride  = gridDim.x * 256;
    float acc = 0.f;
    for (int j = sblk * 256 + threadIdx.x; j < N4; j += stride) {
        float4 p = pred[base + j];
        float4 g = gt[base + j];
        int4   m = mask[base + j];
#pragma unroll
        for (int c = 0; c < 4; ++c) {
            if ((&m.x)[c] != 0) {
                float pc = (&p.x)[c], gc = (&g.x)[c];
                float r  = fabsf(alpha * pc + beta - gc);
                unsigned key = __float_as_uint(r);
                if (mode == 0) {
                    atomicAdd(&h[key >> 16], 1u);
                } else if (mode == 1) {
                    if ((key >> 16) == T) atomicAdd(&h[key & 0xFFFFu], 1u);
                } else {
                    if (key < K) acc += r;
                }
            }
        }
    }
    if (mode == 2) {
        acc = wave_sum_wmma(acc);   // EXEC all-ones at convergence point
        if ((threadIdx.x & 31) == 0) atomicAdd(&dstats[b * 8 + 5], (double)acc);
    }
}

// ---------------------------------------------------------------------------
// Radix-select step over 65536 bins. level 0: find T, cnt_below.
// level 1: find exact key K and tie count n_eq.
// ---------------------------------------------------------------------------
__global__ void __launch_bounds__(256)
k_select(const unsigned* __restrict__ hist, unsigned* __restrict__ prm, int level) {
    const int b = blockIdx.x;
    unsigned* pr = prm + b * 8;
    const unsigned k = pr[2];
    if (k == 0u) return;  // defaults (zeros) leave loss at 0
    const unsigned target = (level == 0) ? k : (k - pr[4]);
    const unsigned* h = hist + (size_t)b * NBINS;
    __shared__ unsigned part[256];
    __shared__ unsigned pref[257];
    const unsigned t = threadIdx.x;
    const unsigned* hseg = h + t * 256;
    unsigned s = 0;
#pragma unroll 8
    for (int i = 0; i < 256; ++i) s += hseg[i];
    part[t] = s;
    __syncthreads();
    if (t == 0) {
        unsigned a = 0;
        for (int i = 0; i < 256; ++i) { pref[i] = a; a += part[i]; }
        pref[256] = a;
    }
    __syncthreads();
    const unsigned lo = pref[t];
    if (lo < target && target <= lo + part[t]) {
        unsigned acc = lo;
        unsigned bin = 255;
        for (int i = 0; i < 256; ++i) {
            unsigned c = hseg[i];
            if (acc + c >= target) { bin = (unsigned)i; break; }
            acc += c;
        }
        unsigned gbin = t * 256 + bin;
        if (level == 0) { pr[3] = gbin; pr[4] = acc; }
        else            { pr[5] = (pr[3] << 16) | gbin; pr[6] = target - acc; }
    }
}

// ---------------------------------------------------------------------------
// Final: per-sample trimmed mean, then mean over batch.
// ---------------------------------------------------------------------------
__global__ void k_final(const double* __restrict__ dstats,
                        const unsigned* __restrict__ prm, float* __restrict__ out) {
    int b = threadIdx.x;
    float loss = 0.f;
    if (b < BATCH) {
        const unsigned* pr = prm + b * 8;
        unsigned k = pr[2];
        float ksum = (float)dstats[b * 8 + 5];
        float kept = ksum + (float)pr[6] * __uint_as_float(pr[5]);
        float mk = kept / fmaxf((float)k, 1.f);
        loss = (k > 0u) ? mk : 0.f;
    }
    loss = wave_sum_shfl(loss);
    if (b == 0) out[0] = loss / (float)BATCH;
}

// ---------------------------------------------------------------------------
extern "C" void kernel_launch(void* const* d_in, const int* in_sizes, int n_in,
                              void* d_out, int out_size, void* d_ws, size_t ws_size,
                              hipStream_t stream) {
    (void)n_in; (void)out_size; (void)ws_size;
    const float4* pred = (const float4*)d_in[0];
    const float4* gt   = (const float4*)d_in[1];
    const int4*   mask = (const int4*)d_in[2];
    float* out = (float*)d_out;

    const int Ntot = in_sizes[0];
    const int N    = Ntot / BATCH;   // 491520
    const int N4   = N / 4;          // 122880

    char* ws = (char*)d_ws;
    double*   dstats = (double*)ws;                 // 32*8 doubles (2048 B): moments + kept_sum
    unsigned* prm    = (unsigned*)(ws + 2048);      // 32*8 uints  (1024 B): alpha,beta,k,T,cnt_below,K,n_eq
    unsigned* hist   = (unsigned*)(ws + 3072);      // 32*65536 uints (8 MB)
    const size_t histBytes = (size_t)BATCH * NBINS * sizeof(unsigned);

    const int bps = 60;                 // blocks per sample: 60*256 = 15360 | 122880
    const dim3 grid(bps, BATCH);        // x: slice, y: sample (no int divide in kernels)

    hipMemsetAsync(ws, 0, 3072, stream);
    k_stats<<<grid, 256, 0, stream>>>(pred, gt, mask, dstats, N4);
    k_params<<<1, 32, 0, stream>>>(dstats, prm);

    hipMemsetAsync(hist, 0, histBytes, stream);
    k_resid<<<grid, 256, 0, stream>>>(pred, gt, mask, prm, hist, dstats, N4, 0);
    k_select<<<BATCH, 256, 0, stream>>>(hist, prm, 0);

    hipMemsetAsync(hist, 0, histBytes, stream);
    k_resid<<<grid, 256, 0, stream>>>(pred, gt, mask, prm, hist, dstats, N4, 1);
    k_select<<<BATCH, 256, 0, stream>>>(hist, prm, 1);

    k_resid<<<grid, 256, 0, stream>>>(pred, gt, mask, prm, hist, dstats, N4, 2);
    k_final<<<1, 32, 0, stream>>>(dstats, prm, out);
}